// BiLSTM_CRF_76020921140162
// MI455X (gfx1250) — compile-verified
//
#include <hip/hip_runtime.h>
#include <stdint.h>

#define KT 11
#define TT 2048
#define BB 4096
#define START_TAG 10
#define STOP_TAG 9
#define STEPS 32                        // timesteps per TDM chunk
#define NCHUNK (TT / STEPS)             // 64
#define ROWQ (STEPS * KT)               // 352 DWORDs of payload per sentence row
#define ROWPAD (ROWQ + ROWQ / 32)       // 363 DWORDs in LDS after TDM padding
#define CHUNK_PAD (16 * ROWPAD)         // 5808 floats per buffer
#define RESC 8                          // rescale period (steps)

typedef float v2f  __attribute__((ext_vector_type(2)));
typedef float v8f  __attribute__((ext_vector_type(8)));
typedef unsigned int u32x2 __attribute__((ext_vector_type(2)));
typedef unsigned int u32x4 __attribute__((ext_vector_type(4)));
typedef int   i32x8 __attribute__((ext_vector_type(8)));
typedef int   i32x4 __attribute__((ext_vector_type(4)));

// TDM load of [16 sentences x 352 floats] with LDS padding: 1 DWORD inserted
// after every 32 DWORDs stored (pad_interval code 4, pad_amount code 0), so
// each 352-DWORD sentence row lands at stride 363 (43 mod 64 -> no bank
// collapse for the emission ds_loads).
__device__ __forceinline__ void tdm_load_tile(uint32_t lds_byte_addr,
                                              const float* gptr) {
  uint64_t ga = (uint64_t)(uintptr_t)gptr;
  u32x4 g0;
  g0[0] = 1u;                                  // count=1 (valid user descriptor)
  g0[1] = lds_byte_addr;                       // lds_addr
  g0[2] = (uint32_t)ga;                        // global_addr[31:0]
  g0[3] = (uint32_t)(ga >> 32) | (2u << 30);   // global_addr[56:32] | type=2
  i32x8 g1;
  const uint32_t TD0 = (uint32_t)ROWQ;         // 352
  const uint32_t STR0 = (uint32_t)(TT * KT);   // 22528
  g1[0] = (int)((2u << 16)                     // data_size = 4B
                | (1u << 20)                   // pad_enable
                | (4u << 22)                   // pad_interval: 32 DWORDs
                | (0u << 25));                 // pad_amount: 1 DWORD
  g1[1] = (int)(TD0 << 16);                    // tensor_dim0[15:0] @ bits 63:48
  g1[2] = (int)((TD0 >> 16) | (16u << 16));    // td0[31:16] | tensor_dim1[15:0]
  g1[3] = (int)(TD0 << 16);                    // td1[31:16]=0 | tile_dim0
  g1[4] = (int)16;                             // tile_dim1=16, tile_dim2=0
  g1[5] = (int)STR0;                           // tensor_dim0_stride[31:0]
  g1[6] = 0;
  g1[7] = 0;
  i32x4 z4 = {0, 0, 0, 0};
  i32x8 z8 = {0, 0, 0, 0, 0, 0, 0, 0};
  __builtin_amdgcn_tensor_load_to_lds(g0, g1, z4, z4, z8, 0);
}

// One v_permlane16_swap_b32 produces both cross-half B slices of a reg pair:
// lo = {a.lanes0-15, b.lanes0-15}, hi = {a.lanes16-31, b.lanes16-31}.
struct FPair { float lo, hi; };
__device__ __forceinline__ FPair plswap(float a, float b) {
  u32x2 r = __builtin_amdgcn_permlane16_swap(__float_as_uint(a),
                                             __float_as_uint(b), false, false);
  FPair p;
  p.lo = __uint_as_float(r[0]);
  p.hi = __uint_as_float(r[1]);
  return p;
}

__global__ __launch_bounds__(32) void crf_forward_kernel(
    const float* __restrict__ feats, const float* __restrict__ trans,
    const float* __restrict__ gold, float* __restrict__ out) {
  __shared__ float lds[2 * CHUNK_PAD + 16];

  const int lane = (int)threadIdx.x;
  const int n = lane & 15;   // sentence column
  const int h = lane >> 4;   // lane half
  const int bBase = (int)blockIdx.x * 16;

  // A operands: A_j[m][k] = exp(trans[m][4j+k]); reg r, lane l ->
  // (m = l%16, k = 4j + r + 2*(l/16)); zero outside KT x KT.
  v2f A[4];
#pragma unroll
  for (int j = 0; j < 4; ++j) {
#pragma unroll
    for (int r = 0; r < 2; ++r) {
      int m = n, k = 4 * j + r + 2 * h;
      float v = 0.0f;
      if (m < KT && k < KT) v = __expf(trans[m * KT + k]);
      A[j][r] = v;
    }
  }

  // exp(trans[STOP][tag]) weights, emission multiplier (2^-5 damping folded
  // in, 0 for padded tags), clamped LDS tag indices.
  float wstop[8], emul[8];
  int tagc[8];
#pragma unroll
  for (int r = 0; r < 8; ++r) {
    int tag = r + 8 * h;
    wstop[r] = (tag < KT) ? __expf(trans[STOP_TAG * KT + tag]) : 0.0f;
    emul[r] = (tag < KT) ? 0.03125f : 0.0f;
    tagc[r] = (tag < KT) ? tag : (KT - 1);
  }

  // alpha init: 1 at START(=10) -> reg2/half1, else 0.
  float d[8];
#pragma unroll
  for (int r = 0; r < 8; ++r) d[r] = 0.0f;
  d[2] = h ? 1.0f : 0.0f;
  float logScale = 0.0f;

  const uint32_t ldsBase = (uint32_t)(uintptr_t)(&lds[0]);
  const float* gbase = feats + (size_t)bBase * TT * KT;
  const int rowOff = n * ROWPAD;

  tdm_load_tile(ldsBase, gbase);

  for (int c = 0; c < NCHUNK; ++c) {
    if (c + 1 < NCHUNK) {
      asm volatile("s_wait_dscnt 0" ::: "memory"); // buffer reads retired
      tdm_load_tile(ldsBase + (uint32_t)(((c + 1) & 1) * CHUNK_PAD * 4),
                    gbase + (size_t)(c + 1) * ROWQ);
      __builtin_amdgcn_s_wait_tensorcnt(1); // chunk c complete (in-order)
    } else {
      __builtin_amdgcn_s_wait_tensorcnt(0);
    }
    const int bufOff = (c & 1) * CHUNK_PAD + rowOff;

    for (int s8 = 0; s8 < STEPS / RESC; ++s8) {
#pragma unroll
      for (int u = 0; u < RESC; ++u) {
        const int s = s8 * RESC + u;
        // D-layout -> B-layout via 4 permlane16_swap
        FPair p0 = plswap(d[0], d[2]);
        FPair p1 = plswap(d[1], d[3]);
        FPair p2 = plswap(d[4], d[6]);
        FPair p3 = plswap(d[5], d[7]);
        v2f B0 = {p0.lo, p1.lo};
        v2f B2 = {p0.hi, p1.hi};
        v2f B1 = {p2.lo, p3.lo};
        v2f B3 = {p2.hi, p3.hi};

        v8f za = {0.f, 0.f, 0.f, 0.f, 0.f, 0.f, 0.f, 0.f};
        v8f zb = {0.f, 0.f, 0.f, 0.f, 0.f, 0.f, 0.f, 0.f};
        v8f acca = __builtin_amdgcn_wmma_f32_16x16x4_f32(
            false, A[0], false, B0, (short)0, za, false, false);
        acca = __builtin_amdgcn_wmma_f32_16x16x4_f32(
            false, A[1], false, B1, (short)0, acca, false, false);
        v8f accb = __builtin_amdgcn_wmma_f32_16x16x4_f32(
            false, A[2], false, B2, (short)0, zb, false, false);
        accb = __builtin_amdgcn_wmma_f32_16x16x4_f32(
            false, A[3], false, B3, (short)0, accb, false, false);

        // emission multiply (independent of the recurrence chain)
        const int cbase = s * KT;
#pragma unroll
        for (int r = 0; r < 8; ++r) {
          const int cc = cbase + tagc[r];
          float e = __expf(lds[bufOff + cc + (cc >> 5)]) * emul[r];
          d[r] = (acca[r] + accb[r]) * e;
        }
      }
      // exact rescale every RESC steps
      float ssum = 0.0f;
#pragma unroll
      for (int r = 0; r < 8; ++r) ssum += d[r];
      ssum += __shfl_xor(ssum, 16, 32);
      logScale += __logf(ssum);
      float inv = 1.0f / ssum;
#pragma unroll
      for (int r = 0; r < 8; ++r) d[r] *= inv;
    }
  }

  // forward = logScale + T*log(32) + log( sum_tag exp(trans[STOP][tag])*alpha )
  float fs = 0.0f;
#pragma unroll
  for (int r = 0; r < 8; ++r) fs += wstop[r] * d[r];
  fs += __shfl_xor(fs, 16, 32);
  const float damp = (float)(TT * 3.4657359027997265); // T * log(32)
  float forward = logScale + damp + __logf(fs);

  if (lane < 16) {
    int b = bBase + n;
    out[b] = forward - gold[b];
  }
}

// Gold path score: one block per sentence, parallel over T.
__global__ __launch_bounds__(256) void crf_gold_kernel(
    const float* __restrict__ feats, const int* __restrict__ tags,
    const float* __restrict__ trans, float* __restrict__ gold) {
  __shared__ float red[256];
  __shared__ float ltr[KT * KT];
  const int b = (int)blockIdx.x;
  const int tid = (int)threadIdx.x;
  if (tid < KT * KT) ltr[tid] = trans[tid];
  __syncthreads();

  const int* trow = tags + (size_t)b * TT;
  const float* frow = feats + (size_t)b * TT * KT;
  float acc = 0.0f;
  for (int t = tid; t < TT; t += 256) {
    int tg = trow[t];
    int pv = (t == 0) ? START_TAG : trow[t - 1];
    acc += frow[(size_t)t * KT + tg] + ltr[tg * KT + pv];
  }
  if (tid == 0) acc += ltr[STOP_TAG * KT + trow[TT - 1]];
  red[tid] = acc;
  __syncthreads();
  for (int off = 128; off > 0; off >>= 1) {
    if (tid < off) red[tid] += red[tid + off];
    __syncthreads();
  }
  if (tid == 0) gold[b] = red[0];
}

extern "C" void kernel_launch(void* const* d_in, const int* in_sizes, int n_in,
                              void* d_out, int out_size, void* d_ws,
                              size_t ws_size, hipStream_t stream) {
  (void)in_sizes; (void)n_in; (void)out_size; (void)ws_size;
  const float* feats = (const float*)d_in[0];
  const int*   tags  = (const int*)d_in[1];
  const float* trans = (const float*)d_in[2];
  float* out  = (float*)d_out;
  float* gold = (float*)d_ws; // BB floats of scratch

  crf_gold_kernel<<<BB, 256, 0, stream>>>(feats, tags, trans, gold);
  crf_forward_kernel<<<BB / 16, 32, 0, stream>>>(feats, trans, gold, out);
}